// CentroidLoss_35373350650036
// MI455X (gfx1250) — compile-verified
//
#include <hip/hip_runtime.h>

// CentroidLoss fused kernel for MI455X (gfx1250, wave32).
//  - One streaming pass: BCE (channels 0-2) + centroid-similarity gather.
//  - Cross-lane reduction via V_WMMA_F32_16X16X4_F32 (A = lane partials,
//    B = ones -> grand total; layout-agnostic).
//  - Deterministic two-stage reduction (fixed-slot partials, fixed-order final sum).

typedef __attribute__((ext_vector_type(2))) float v2f;
typedef __attribute__((ext_vector_type(8))) float v8f;

#define N_VOX  819200   // 8*320*320
#define SLICE  102400   // 320*320
#define ROW    320
#define NEPS   1e-7f

// Wave32 sum via f32 WMMA: D = A(16x4) * Ones(4x16). Each lane contributes its
// value in A-fragment slot 0 (slot 1 = 0). Sum of a lane's 8 D registers covers
// rows {0..7} (lanes 0-15) or {8..15} (lanes 16-31); xor-16 combines halves.
__device__ __forceinline__ float wave_sum_wmma(float v) {
  v2f a; a[0] = v;    a[1] = 0.0f;
  v2f b; b[0] = 1.0f; b[1] = 1.0f;
  v8f c = {};
  c = __builtin_amdgcn_wmma_f32_16x16x4_f32(false, a, false, b,
                                            (short)0, c, false, false);
  float r = ((c[0] + c[1]) + (c[2] + c[3])) + ((c[4] + c[5]) + (c[6] + c[7]));
  r += __shfl_xor(r, 16);
  return r;   // total across all 32 lanes, broadcast to every lane
}

__device__ __forceinline__ float bce_term(float xv, float tv) {
  float p = fminf(fmaxf(xv, NEPS), 1.0f - NEPS);
  return -(tv * __logf(p) + (1.0f - tv) * __logf(1.0f - p));
}

__global__ __launch_bounds__(256) void
centloss_main(const float* __restrict__ x, const float* __restrict__ t,
              float* __restrict__ partials) {
  const int tid = blockIdx.x * 256 + threadIdx.x;
  const int i0  = tid * 4;            // grid sized so i0+3 < N_VOX exactly

  // ---- streaming loads (b128) ----
  const float4 x0 = *(const float4*)(x + i0);
  const float4 x1 = *(const float4*)(x + N_VOX + i0);
  const float4 x2 = *(const float4*)(x + 2 * N_VOX + i0);
  const float4 t0 = *(const float4*)(t + i0);
  const float4 t1 = *(const float4*)(t + N_VOX + i0);
  const float4 t2 = *(const float4*)(t + 2 * N_VOX + i0);
  const float4 t3 = *(const float4*)(t + 3 * N_VOX + i0);

  // ---- BCE partial sums per channel ----
  float s0 = bce_term(x0.x, t0.x) + bce_term(x0.y, t0.y) +
             bce_term(x0.z, t0.z) + bce_term(x0.w, t0.w);
  float s1 = bce_term(x1.x, t1.x) + bce_term(x1.y, t1.y) +
             bce_term(x1.z, t1.z) + bce_term(x1.w, t1.w);
  float s2 = bce_term(x2.x, t2.x) + bce_term(x2.y, t2.y) +
             bce_term(x2.z, t2.z) + bce_term(x2.w, t2.w);

  // ---- centroid similarity (rare path: ~1e-4 of voxels) ----
  float csum[4] = {0.0f, 0.0f, 0.0f, 0.0f};
  float cc = 0.0f;
  const float tcs[4] = {t3.x, t3.y, t3.z, t3.w};
  const float inv_dmax = 0.17407765595569785f;   // 1/sqrt(33)
  for (int e = 0; e < 4; ++e) {
    if (tcs[e] > 0.5f) {
      const int vi = i0 + e;
      float g0 = 0.0f, g1 = 0.0f, g2 = 0.0f, g3 = 0.0f, cnt = 0.0f;
      for (int k = 0; k < 243; ++k) {           // same C-order as np.nonzero
        const int dz  = k / 81 - 1;
        const int r81 = k % 81;
        const int dy  = r81 / 9 - 4;
        const int dx  = r81 % 9 - 4;
        const int ni  = vi + dz * SLICE + dy * ROW + dx;
        if (ni >= 0 && ni < N_VOX) {
          const float w =
              sqrtf((float)(dz * dz + dy * dy + dx * dx)) * inv_dmax - 1.0f;
          g0 += x[ni] * w;
          g1 += x[N_VOX + ni] * w;
          g2 += x[2 * N_VOX + ni] * w;
          g3 += x[3 * N_VOX + ni] * w;
          cnt += 1.0f;
        }
      }
      const float inv = 1.0f / fmaxf(cnt, 1.0f);
      csum[0] += g0 * inv; csum[1] += g1 * inv;
      csum[2] += g2 * inv; csum[3] += g3 * inv;
      cc += 1.0f;
    }
  }

  // ---- hierarchical reduction: WMMA (wave) -> LDS (block) ----
  float vals[8] = {s0, s1, s2, csum[0], csum[1], csum[2], csum[3], cc};
  float r[8];
#pragma unroll
  for (int j = 0; j < 8; ++j) r[j] = wave_sum_wmma(vals[j]);  // EXEC all-1s here

  __shared__ float red[64];                  // 8 waves x 8 components
  const int wave = threadIdx.x >> 5;
  const int lane = threadIdx.x & 31;
  if (lane == 0) {
#pragma unroll
    for (int j = 0; j < 8; ++j) red[wave * 8 + j] = r[j];
  }
  __syncthreads();
  if (threadIdx.x == 0) {
#pragma unroll
    for (int j = 0; j < 8; ++j) {
      float a = 0.0f;
      for (int w = 0; w < 8; ++w) a += red[w * 8 + j];
      partials[blockIdx.x * 8 + j] = a;      // fixed slot -> deterministic
    }
  }
}

__global__ __launch_bounds__(32) void
centloss_finalize(const float* __restrict__ partials, float* __restrict__ out,
                  int nblocks) {
  __shared__ float S[8];
  if (threadIdx.x < 8) {
    float a = 0.0f;
    for (int b = 0; b < nblocks; ++b)        // fixed order -> deterministic
      a += partials[b * 8 + threadIdx.x];
    S[threadIdx.x] = a;
  }
  __syncthreads();
  if (threadIdx.x == 0) {
    const float loss  = (S[0] + S[1] + S[2]) / (3.0f * (float)N_VOX);
    const float ncent = fmaxf(S[7], 1.0f);
    const float sim0 = S[3] / ncent, sim1 = S[4] / ncent;
    const float sim2 = S[5] / ncent, sim3 = S[6] / ncent;
    const float aff_pen  = (sim0 + sim1 + sim2) * (1.0f / 3.0f) * 0.5f; // PHI
    const float cent_pen = (1.0f - sim3) * 0.5f;                        // ETA
    out[0] = loss + aff_pen + cent_pen;
  }
}

extern "C" void kernel_launch(void* const* d_in, const int* in_sizes, int n_in,
                              void* d_out, int out_size, void* d_ws, size_t ws_size,
                              hipStream_t stream) {
  const float* x = (const float*)d_in[0];   // inputs  (1,4,8,320,320) f32
  const float* t = (const float*)d_in[1];   // targets (1,4,8,320,320) f32
  float* partials = (float*)d_ws;           // 800*8 f32 = 25.6 KB scratch
  float* out      = (float*)d_out;          // 1 f32

  const int nblocks = N_VOX / (256 * 4);    // = 800, exact cover, no tail
  centloss_main<<<nblocks, 256, 0, stream>>>(x, t, partials);
  centloss_finalize<<<1, 32, 0, stream>>>(partials, out, nblocks);
}